// GCNHeterogeneous_89584427860363
// MI455X (gfx1250) — compile-verified
//
#include <hip/hip_runtime.h>

#define NN 50000   // nodes
#define EE 800000  // edges per relation
#define RR 2       // relations
#define DD 128     // channels
#define GG 64      // graphs
#define CC 8       // classes

typedef __attribute__((ext_vector_type(2))) float v2f;
typedef __attribute__((ext_vector_type(8))) float v8f;

__global__ void zero_f32(float* __restrict__ p, int n) {
    int i = blockIdx.x * blockDim.x + threadIdx.x;
    if (i < n) p[i] = 0.0f;
}

// deg[r*N + dst] += 1 over all relations' edges
__global__ void deg_accum(const int* __restrict__ ei, float* __restrict__ deg) {
    int i = blockIdx.x * blockDim.x + threadIdx.x;  // over R*E
    if (i >= RR * EE) return;
    int r = i / EE;
    int e = i - r * EE;
    int dst = ei[(size_t)r * 2 * EE + EE + e];
    atomicAdd(&deg[(size_t)r * NN + dst], 1.0f);
}

__global__ void deg_finalize(float* __restrict__ isd) {
    int i = blockIdx.x * blockDim.x + threadIdx.x;
    if (i < RR * NN) isd[i] = rsqrtf(isd[i] + 1.0f);
}

// xw = H @ Wm ; out += xw * isd^2 + bias    (fused GCNConv self-loop + bias)
// One block = 16 rows x 128 cols. 8 waves, each owns a 16x16 tile via
// V_WMMA_F32_16X16X4_F32 chained over K=128.
__global__ __launch_bounds__(256) void gcn_gemm(
    const float* __restrict__ H, const float* __restrict__ Wm,
    const float* __restrict__ bias, const float* __restrict__ isd,
    float* __restrict__ xw, float* __restrict__ out)
{
    __shared__ float tileA[16 * 132];  // padded stride: avoid LDS bank conflicts
    // cooperative coalesced load of 16x128 fp32 A tile
    const float4* s4 = (const float4*)(H + (size_t)blockIdx.x * (16 * DD));
    for (int j = threadIdx.x; j < 512; j += 256) {
        int row = j >> 5;   // 32 float4 per source row
        int c4  = j & 31;
        ((float4*)tileA)[row * 33 + c4] = s4[j];
    }
    __syncthreads();

    const int lane = threadIdx.x & 31;
    const int wave = threadIdx.x >> 5;
    const int lrow = lane & 15;
    const int half = lane >> 4;
    const int col  = wave * 16 + lrow;   // N index for B/C/D fragments

    v8f c = {};
    #pragma unroll 8
    for (int k = 0; k < DD; k += 4) {
        int ka = k + 2 * half;                    // K of VGPR0 for this lane-half
        v2f a = *(const v2f*)&tileA[lrow * 132 + ka];  // A[M=lrow][ka, ka+1]
        v2f b;
        b.x = Wm[(ka + 0) * DD + col];            // B[K=ka  ][N=col]
        b.y = Wm[(ka + 1) * DD + col];            // B[K=ka+1][N=col]
        c = __builtin_amdgcn_wmma_f32_16x16x4_f32(
                false, a, false, b, (short)0, c, false, false);
    }

    const int row0 = blockIdx.x * 16;
    const float bcol = bias[col];
    #pragma unroll
    for (int v = 0; v < 8; ++v) {
        int row = row0 + v + 8 * half;            // D layout: VGPR v -> M = v + 8*half
        float val = c[v];
        size_t idx = (size_t)row * DD + col;
        xw[idx] = val;
        float s = isd[row];
        out[idx] += val * (s * s) + bcol;         // self-loop msg + bias
    }
}

// out[dst] += xw[src] * isd[src]*isd[dst]   (one wave = 32 channels of one edge)
__global__ void gcn_scatter(const float* __restrict__ xw, const int* __restrict__ srcI,
                            const int* __restrict__ dstI, const float* __restrict__ isd,
                            float* __restrict__ out)
{
    int d = threadIdx.x;  // 0..127
    int slot = blockIdx.x * blockDim.y + threadIdx.y;
    int nslots = gridDim.x * blockDim.y;
    for (int e = slot; e < EE; e += nslots) {
        int s = srcI[e], t = dstI[e];
        float norm = isd[s] * isd[t];
        atomicAdd(&out[(size_t)t * DD + d], xw[(size_t)s * DD + d] * norm);
    }
}

__global__ void relu_inplace(float* __restrict__ p, int n) {
    int i = blockIdx.x * blockDim.x + threadIdx.x;
    if (i < n) p[i] = fmaxf(p[i], 0.0f);
}

__global__ void count_nodes(const int* __restrict__ batch, float* __restrict__ cnts) {
    int i = blockIdx.x * blockDim.x + threadIdx.x;
    if (i < NN) atomicAdd(&cnts[batch[i]], 1.0f);
}

// per-block LDS partial sums: thread d exclusively owns channel d -> no LDS atomics
__global__ __launch_bounds__(128) void pool_sums(
    const float* __restrict__ h, const int* __restrict__ batch, float* __restrict__ sums)
{
    __shared__ float acc[GG * DD];  // 32 KB
    int d = threadIdx.x;
    for (int g = 0; g < GG; ++g) acc[g * DD + d] = 0.0f;
    int per = (NN + gridDim.x - 1) / gridDim.x;
    int start = blockIdx.x * per;
    int end = start + per; if (end > NN) end = NN;
    for (int i = start; i < end; ++i) {
        int g = batch[i];
        acc[g * DD + d] += h[(size_t)i * DD + d];
    }
    for (int g = 0; g < GG; ++g) atomicAdd(&sums[g * DD + d], acc[g * DD + d]);
}

__global__ void final_linear(const float* __restrict__ sums, const float* __restrict__ cnts,
                             const float* __restrict__ lin_w, const float* __restrict__ lin_b,
                             float* __restrict__ out)
{
    int t = threadIdx.x;             // 0..511
    int g = t >> 3, c = t & 7;
    float inv = 1.0f / fmaxf(cnts[g], 1.0f);
    float acc = lin_b[c];
    for (int d = 0; d < DD; ++d)
        acc += sums[g * DD + d] * inv * lin_w[d * CC + c];
    out[(size_t)g * CC + c] = acc;
}

extern "C" void kernel_launch(void* const* d_in, const int* in_sizes, int n_in,
                              void* d_out, int out_size, void* d_ws, size_t ws_size,
                              hipStream_t stream) {
    const float* x     = (const float*)d_in[0];   // [N, D]
    const float* W     = (const float*)d_in[1];   // [L, R, D, D]
    const float* b     = (const float*)d_in[2];   // [L, R, D]
    const float* lin_w = (const float*)d_in[3];   // [D, C]
    const float* lin_b = (const float*)d_in[4];   // [C]
    const int*   ei    = (const int*)d_in[5];     // [R, 2, E]
    const int*   batch = (const int*)d_in[6];     // [N]
    float* out = (float*)d_out;

    const size_t ND = (size_t)NN * DD;
    float* ws   = (float*)d_ws;
    float* bufA = ws;                        // layer0 out -> layer1 h
    float* bufC = bufA + ND;                 // layer1 out
    float* isd  = bufC + ND;                 // [R, N]  deg -> rsqrt(deg+1)
    float* sums = isd + (size_t)RR * NN;     // [G, D]
    float* cnts = sums + (size_t)GG * DD;    // [G]
    float* bufB = cnts + GG;                 // xw scratch (fully overwritten each GEMM)

    // zero contiguous [bufA, bufC, isd, sums, cnts]
    int zcount = (int)(2 * ND + (size_t)RR * NN + (size_t)GG * DD + GG);
    zero_f32<<<(zcount + 255) / 256, 256, 0, stream>>>(bufA, zcount);

    deg_accum<<<(RR * EE + 255) / 256, 256, 0, stream>>>(ei, isd);
    deg_finalize<<<(RR * NN + 255) / 256, 256, 0, stream>>>(isd);

    dim3 sblk(128, 2);
    const int sgrid = 4096;

    // layer 0: h = x, accumulate into bufA
    for (int r = 0; r < RR; ++r) {
        const float* Wlr = W + (size_t)r * DD * DD;
        const float* blr = b + (size_t)r * DD;
        const float* isd_r = isd + (size_t)r * NN;
        const int* src = ei + (size_t)r * 2 * EE;
        const int* dst = src + EE;
        gcn_gemm<<<NN / 16, 256, 0, stream>>>(x, Wlr, blr, isd_r, bufB, bufA);
        gcn_scatter<<<sgrid, sblk, 0, stream>>>(bufB, src, dst, isd_r, bufA);
    }
    relu_inplace<<<((int)ND + 255) / 256, 256, 0, stream>>>(bufA, (int)ND);

    // layer 1: h = bufA, accumulate into bufC
    for (int r = 0; r < RR; ++r) {
        const float* Wlr = W + (size_t)(RR + r) * DD * DD;
        const float* blr = b + (size_t)(RR + r) * DD;
        const float* isd_r = isd + (size_t)r * NN;
        const int* src = ei + (size_t)r * 2 * EE;
        const int* dst = src + EE;
        gcn_gemm<<<NN / 16, 256, 0, stream>>>(bufA, Wlr, blr, isd_r, bufB, bufC);
        gcn_scatter<<<sgrid, sblk, 0, stream>>>(bufB, src, dst, isd_r, bufC);
    }

    count_nodes<<<(NN + 255) / 256, 256, 0, stream>>>(batch, cnts);
    pool_sums<<<64, 128, 0, stream>>>(bufC, batch, sums);
    final_linear<<<1, GG * CC, 0, stream>>>(sums, cnts, lin_w, lin_b, out);
}